// NuclearLattice_58153857188398
// MI455X (gfx1250) — compile-verified
//
#include <hip/hip_runtime.h>
#include <hip/hip_bf16.h>

typedef __attribute__((ext_vector_type(2))) float v2f;
typedef __attribute__((ext_vector_type(8))) float v8f;

#define ATOMS 6144
#define TILES 384           // 6144 / 16
#define WAVES_PER_BLOCK 8   // 256 threads = 8 wave32

// ---------------------------------------------------------------- init
__global__ void nl_init(double* acc, unsigned* flag) {
    if (threadIdx.x == 0) { acc[0] = 0.0; flag[0] = 0u; }
}

// ------------------------------------------------------- bounds check
// blocked_b: any of first 4 state columns outside [lo, hi]
__global__ void nl_bounds(const float* __restrict__ states, unsigned* __restrict__ flag) {
    int a = blockIdx.x * blockDim.x + threadIdx.x;
    if (a >= ATOMS) return;
    const float* s = states + a * 5;
    bool bad = false;
    bad |= (s[0] < -32.0f) || (s[0] > 32.0f);
    bad |= (s[1] < -32.0f) || (s[1] > 32.0f);
    bad |= (s[2] < -32.0f) || (s[2] > 32.0f);
    bad |= (s[3] < -0.5f)  || (s[3] > 0.5f);
    if (bad) atomicOr(flag, 1u);
}

// ---------------------------------------------------- pairwise energy
// One wave per 16x16 tile of the pair matrix. Two f32 WMMAs build the
// position Gram (K = x,y,z,0) and, accumulated on top, the full 5-D
// state Gram (K += spin,iso,0,0).
__global__ __launch_bounds__(256) void nl_pair(const float* __restrict__ states,
                                               double* __restrict__ acc,
                                               unsigned* __restrict__ flag) {
    const int lane = threadIdx.x & 31;
    const int wave = threadIdx.x >> 5;
    const int tj = blockIdx.x * WAVES_PER_BLOCK + wave;
    const int ti = blockIdx.y;
    if (tj < ti) return;                       // uniform per wave: EXEC stays all-1

    const int ibase = ti * 16;
    const int jbase = tj * 16;
    const int l  = lane & 15;
    const bool hi = lane >= 16;                // lanes 16..31 carry K=2,3

    // ---- A fragment: rows ibase..ibase+15  (32-bit A 16x4 ISA layout)
    const float* Sa = states + (ibase + l) * 5;
    float ax = Sa[0], ay = Sa[1], az = Sa[2], as = Sa[3], ai = Sa[4];
    v2f a1, a2;
    a1[0] = hi ? az   : ax;   a1[1] = hi ? 0.0f : ay;   // K = (x,y | z,0)
    a2[0] = hi ? 0.0f : as;   a2[1] = hi ? 0.0f : ai;   // K = (spin,iso | 0,0)

    // ---- B fragment: rows jbase..jbase+15 (same striping, B = A-blockᵀ)
    const float* Sb = states + (jbase + l) * 5;
    float bx = Sb[0], by = Sb[1], bz = Sb[2], bs = Sb[3], bi = Sb[4];
    v2f b1, b2;
    b1[0] = hi ? bz   : bx;   b1[1] = hi ? 0.0f : by;
    b2[0] = hi ? 0.0f : bs;   b2[1] = hi ? 0.0f : bi;

    v8f cz = {};
    // position Gram: P · Pᵀ  (16x16 f32)
    v8f gpos = __builtin_amdgcn_wmma_f32_16x16x4_f32(
        false, a1, false, b1, (short)0, cz, false, false);
    // full 5-D state Gram = gpos + spin·spin + iso·iso (accumulate)
    v8f gst  = __builtin_amdgcn_wmma_f32_16x16x4_f32(
        false, a2, false, b2, (short)0, gpos, false, false);

    // ---- epilogue: per-lane column j, per-VGPR row i = ibase + v (+8 hi)
    const int j = jbase + l;
    const float* Sj = states + j * 5;
    float xj = Sj[0], yj = Sj[1], zj = Sj[2], spj = Sj[3], isj = Sj[4];
    float psq_j = xj*xj + yj*yj + zj*zj;
    float sq_j  = psq_j + spj*spj + isj*isj;
    float qj    = isj + 0.5f;

    const float wgt = (ti == tj) ? 0.5f : 1.0f;
    float esum = 0.0f;
    bool  blocked = false;

#pragma unroll
    for (int v = 0; v < 8; ++v) {
        int i = ibase + v + (hi ? 8 : 0);
        const float* Si = states + i * 5;
        float xi = Si[0], yi = Si[1], zi = Si[2], spi = Si[3], isi = Si[4];
        float psq_i = xi*xi + yi*yi + zi*zi;
        float sq_i  = psq_i + spi*spi + isi*isi;
        float qi    = isi + 0.5f;

        float d2  = fmaxf(psq_i + psq_j - 2.0f * gpos[v], 0.0f);
        float d2s = fmaxf(sq_i  + sq_j  - 2.0f * gst[v],  0.0f);
        bool diag = (i == j);
        blocked |= (!diag) && (d2s < 1e-6f);

        float gauss = __expf(-0.25f * d2);                 // 1/A_RANGE^2 = 0.25
        float e = (-10.0f + 2.0f * (spi * spj)) * gauss
                + 1.44f * (qi * qj) * rsqrtf(d2 + 1e-6f);
        esum += diag ? 0.0f : e;
    }
    esum *= wgt;

    // wave32 reduction
#pragma unroll
    for (int o = 16; o > 0; o >>= 1)
        esum += __shfl_xor(esum, o, 32);

    if (blocked) atomicOr(flag, 2u);        // essentially never taken
    if (lane == 0) atomicAdd(acc, (double)esum);
}

// ------------------------------------------------------------ finalize
__global__ void nl_finalize(const double* acc, const unsigned* flag, float* out) {
    if (threadIdx.x == 0)
        out[0] = flag[0] ? __builtin_inff() : (float)acc[0];
}

// ------------------------------------------------------------- launch
extern "C" void kernel_launch(void* const* d_in, const int* in_sizes, int n_in,
                              void* d_out, int out_size, void* d_ws, size_t ws_size,
                              hipStream_t stream) {
    const float* states = (const float*)d_in[0];
    double*   acc  = (double*)d_ws;
    unsigned* flag = (unsigned*)((char*)d_ws + 8);
    float*    out  = (float*)d_out;

    nl_init<<<1, 32, 0, stream>>>(acc, flag);
    nl_bounds<<<(ATOMS + 255) / 256, 256, 0, stream>>>(states, flag);
    dim3 grid(TILES / WAVES_PER_BLOCK, TILES);   // 48 x 384 blocks, 8 waves each
    nl_pair<<<grid, 256, 0, stream>>>(states, acc, flag);
    nl_finalize<<<1, 32, 0, stream>>>(acc, flag, out);
}